// CoreRNNFW_81441169867232
// MI455X (gfx1250) — compile-verified
//
#include <hip/hip_runtime.h>
#include <math.h>

typedef __attribute__((ext_vector_type(2))) float v2f;
typedef __attribute__((ext_vector_type(8))) float v8f;

#define D_G 256
#define D_H 512
#define D_OUT 256
#define BSZ 64
#define TSTEPS 16
#define LAMBDA 0.95f
#define ETA 0.5f
#define LN_EPS 1e-5f

// ---------- reductions (wave32) ----------
__device__ inline float wave_sum(float v) {
  #pragma unroll
  for (int m = 16; m >= 1; m >>= 1) v += __shfl_xor(v, m, 32);
  return v;
}

template <int NWAVES>
__device__ inline float block_sum(float v, float* scratch) {
  const int lane = threadIdx.x & 31;
  const int wid  = threadIdx.x >> 5;
  v = wave_sum(v);
  if (lane == 0) scratch[wid] = v;
  __syncthreads();
  if (wid == 0) {
    float x = (lane < NWAVES) ? scratch[lane] : 0.0f;
    x = wave_sum(x);
    if (lane == 0) scratch[0] = x;
  }
  __syncthreads();
  float r = scratch[0];
  __syncthreads();
  return r;
}

// ---------- fused GEMM (WMMA f32) + bias + LN + ReLU ----------
// h_base = h @ Wh^T + z @ Wg^T + bh ;  h_s = relu(LN(h_base))
// grid = 4 blocks (16-row M stripes), block = 512 threads (16 waves).
// Wave w computes two 16x16 N-tiles at n0 = 32*w, k-loop via V_WMMA_F32_16X16X4_F32.
__global__ __launch_bounds__(512) void gemm_base_kernel(
    const float* __restrict__ h,   // [64,512] (ignored if use_h == 0)
    const float* __restrict__ z,   // [64,256]
    const float* __restrict__ Wh,  // [512,512]
    const float* __restrict__ Wg,  // [512,256]
    const float* __restrict__ bh,  // [512]
    const float* __restrict__ lng,
    const float* __restrict__ lnb,
    float* __restrict__ h_base,    // [64,512]
    float* __restrict__ h_s,       // [64,512] (may alias h; block-local rows only)
    int use_h)
{
  __shared__ float tile[16][D_H];  // 32 KB
  const int m0   = blockIdx.x * 16;
  const int lane = threadIdx.x & 31;
  const int wid  = threadIdx.x >> 5;   // 0..15
  const int n0   = wid * 32;           // two N tiles: n0, n0+16
  const int mrow = lane & 15;          // A fragment: M row
  const int kk   = (lane >> 4) * 2;    // A/B fragment: K sub-offset (lanes 16-31 -> K+2)
  const int ncol = lane & 15;          // B/D fragment: N col

  v8f acc0 = {};
  v8f acc1 = {};

  if (use_h) {
    const float* hrow = h + (m0 + mrow) * D_H;
    for (int k = 0; k < D_H; k += 4) {
      v2f a;  a.x = hrow[k + kk];                    a.y = hrow[k + kk + 1];
      v2f b0; b0.x = Wh[(n0 + ncol) * D_H + k + kk]; b0.y = Wh[(n0 + ncol) * D_H + k + kk + 1];
      v2f b1; b1.x = Wh[(n0 + 16 + ncol) * D_H + k + kk];
              b1.y = Wh[(n0 + 16 + ncol) * D_H + k + kk + 1];
      acc0 = __builtin_amdgcn_wmma_f32_16x16x4_f32(false, a, false, b0, (short)0, acc0, false, false);
      acc1 = __builtin_amdgcn_wmma_f32_16x16x4_f32(false, a, false, b1, (short)0, acc1, false, false);
    }
  }
  {
    const float* zrow = z + (m0 + mrow) * D_G;
    for (int k = 0; k < D_G; k += 4) {
      v2f a;  a.x = zrow[k + kk];                    a.y = zrow[k + kk + 1];
      v2f b0; b0.x = Wg[(n0 + ncol) * D_G + k + kk]; b0.y = Wg[(n0 + ncol) * D_G + k + kk + 1];
      v2f b1; b1.x = Wg[(n0 + 16 + ncol) * D_G + k + kk];
              b1.y = Wg[(n0 + 16 + ncol) * D_G + k + kk + 1];
      acc0 = __builtin_amdgcn_wmma_f32_16x16x4_f32(false, a, false, b0, (short)0, acc0, false, false);
      acc1 = __builtin_amdgcn_wmma_f32_16x16x4_f32(false, a, false, b1, (short)0, acc1, false, false);
    }
  }

  // D layout: VGPR i -> row (lane>>4)*8 + i, col = lane&15
  const int rbase = (lane >> 4) * 8;
  #pragma unroll
  for (int i = 0; i < 8; i++) {
    tile[rbase + i][n0 + ncol]      = acc0[i];
    tile[rbase + i][n0 + 16 + ncol] = acc1[i];
  }
  __syncthreads();

  // LayerNorm: wave `wid` owns row `wid` (16 waves == 16 rows)
  float vals[16];
  float sum = 0.0f, sq = 0.0f;
  #pragma unroll
  for (int j = 0; j < 16; j++) {
    const int c = lane + 32 * j;
    float x = tile[wid][c] + bh[c];
    vals[j] = x;
    sum += x; sq += x * x;
  }
  sum = wave_sum(sum);
  sq  = wave_sum(sq);
  const float mu  = sum * (1.0f / D_H);
  const float var = sq * (1.0f / D_H) - mu * mu;
  const float rs  = rsqrtf(var + LN_EPS);
  #pragma unroll
  for (int j = 0; j < 16; j++) {
    const int c = lane + 32 * j;
    const float xb = vals[j];
    const float y  = (xb - mu) * rs * lng[c] + lnb[c];
    h_base[(m0 + wid) * D_H + c] = xb;
    h_s[(m0 + wid) * D_H + c]    = fmaxf(y, 0.0f);
  }
}

// ---------- fast-weight settle: A·h via history of rank-1 terms ----------
// A_{t-1}·h = sum_{s<t} eta*lambda^(t-1-s) * (hist_s · h) * hist_s
// grid = 64 blocks (one per batch element), block = 512 threads.
__global__ __launch_bounds__(512) void settle_kernel(
    float* __restrict__ h_state,        // in: h_s from gemm; out: settled h
    const float* __restrict__ h_base,   // [64,512]
    float* __restrict__ hist,           // [15,64,512]
    const float* __restrict__ lng,
    const float* __restrict__ lnb,
    const float* __restrict__ alpha_ptr,
    int t, int is_final)
{
  __shared__ float hs[D_H];
  __shared__ float hb[D_H];
  __shared__ float histl[15 * D_H];   // 30 KB history cache (fits 320 KB WGP LDS easily)
  __shared__ float dots[16];
  __shared__ float scratch[16];

  const int b    = blockIdx.x;
  const int tid  = threadIdx.x;
  const int lane = tid & 31;
  const int wid  = tid >> 5;

  hs[tid] = h_state[b * D_H + tid];
  hb[tid] = h_base[b * D_H + tid];
  for (int s = 0; s < t; s++)
    histl[s * D_H + tid] = hist[(s * BSZ + b) * D_H + tid];
  __syncthreads();

  const float alpha = alpha_ptr[0];
  const float kpow  = (alpha >= 0.0f) ? (1.0f + log1pf(expf(alpha)))
                                      : (1.0f / (1.0f + log1pf(expf(-alpha))));

  for (int it = 0; it < 3; it++) {
    // one wave per history entry computes its dot product (t <= 15 <= 16 waves)
    if (wid < t) {
      float d = 0.0f;
      #pragma unroll
      for (int j = 0; j < 16; j++) {
        const int c = lane + 32 * j;
        d += histl[wid * D_H + c] * hs[c];
      }
      d = wave_sum(d);
      if (lane == 0) dots[wid] = d;
    }
    __syncthreads();

    float ahi = 0.0f;
    for (int s = 0; s < t; s++) {
      const float c = ETA * powf(LAMBDA, (float)(t - 1 - s));
      ahi += c * dots[s] * histl[s * D_H + tid];
    }
    const float hsi = hs[tid];

    float newv;
    if (!is_final) {
      const float d_ha = block_sum<16>(hsi * ahi, scratch);
      const float n1sq = block_sum<16>(hsi * hsi, scratch);
      const float n2sq = block_sum<16>(ahi * ahi, scratch);
      const float n1 = fmaxf(sqrtf(n1sq), 1e-6f);
      const float n2 = fmaxf(sqrtf(n2sq), 1e-6f);
      float R = d_ha / (n1 * n2);
      R = fminf(fmaxf(R, 0.0f), 1.0f);
      const float a = 1.0f - powf(1.0f - R, kpow);
      newv = (1.0f - a * a) * hb[tid] + a * ahi;
    } else {
      newv = hb[tid] + ahi;
    }

    const float mu  = block_sum<16>(newv, scratch) * (1.0f / D_H);
    const float msq = block_sum<16>(newv * newv, scratch) * (1.0f / D_H);
    const float var = msq - mu * mu;
    float y = (newv - mu) * rsqrtf(var + LN_EPS) * lng[tid] + lnb[tid];
    y = fmaxf(y, 0.0f);
    __syncthreads();
    hs[tid] = y;
    __syncthreads();
  }

  h_state[b * D_H + tid] = hs[tid];
  if (!is_final) hist[(t * BSZ + b) * D_H + tid] = hs[tid];
}

// ---------- head GEMM (WMMA f32): pred = h @ W_head^T + b_head ----------
// grid = 4 blocks (M stripes), block = 512 threads, wave w -> N tile n0 = 16*w.
__global__ __launch_bounds__(512) void head_kernel(
    const float* __restrict__ h,      // [64,512]
    const float* __restrict__ Whead,  // [256,512]
    const float* __restrict__ bhead,  // [256]
    float* __restrict__ pred)         // [64,256]
{
  const int m0   = blockIdx.x * 16;
  const int lane = threadIdx.x & 31;
  const int wid  = threadIdx.x >> 5;
  const int n0   = wid * 16;
  const int mrow = lane & 15;
  const int kk   = (lane >> 4) * 2;
  const int ncol = lane & 15;

  v8f acc = {};
  const float* hrow = h + (m0 + mrow) * D_H;
  const float* wrow = Whead + (n0 + ncol) * D_H;
  for (int k = 0; k < D_H; k += 4) {
    v2f a;  a.x = hrow[k + kk]; a.y = hrow[k + kk + 1];
    v2f bb; bb.x = wrow[k + kk]; bb.y = wrow[k + kk + 1];
    acc = __builtin_amdgcn_wmma_f32_16x16x4_f32(false, a, false, bb, (short)0, acc, false, false);
  }
  const int rbase = (lane >> 4) * 8;
  const float bias = bhead[n0 + ncol];
  #pragma unroll
  for (int i = 0; i < 8; i++)
    pred[(m0 + rbase + i) * D_OUT + n0 + ncol] = acc[i] + bias;
}

// ---------- loss/cosine ----------
__global__ __launch_bounds__(256) void loss_kernel(
    const float* __restrict__ pred,   // [64,256]
    const float* __restrict__ clean,  // [64,256]
    float* __restrict__ out)          // [2]
{
  __shared__ float scratch[8];
  const int tid = threadIdx.x;  // == column, 256 threads
  float acc_sq = 0.0f, acc_dot = 0.0f;
  for (int b = 0; b < BSZ; b++) {
    const float p = pred[b * D_OUT + tid];
    const float c = clean[b * D_OUT + tid];
    const float np = block_sum<8>(p * p, scratch);
    const float nc = block_sum<8>(c * c, scratch);
    const float ph = p / (sqrtf(np) + 1e-6f);
    const float ch = c / (sqrtf(nc) + 1e-6f);
    acc_sq  += (ph - ch) * (ph - ch);
    acc_dot += ph * ch;
  }
  const float tsq  = block_sum<8>(acc_sq, scratch);
  const float tdot = block_sum<8>(acc_dot, scratch);
  if (tid == 0) {
    out[0] = tsq * (1.0f / (BSZ * D_OUT));
    out[1] = tdot * (1.0f / BSZ);
  }
}

extern "C" void kernel_launch(void* const* d_in, const int* in_sizes, int n_in,
                              void* d_out, int out_size, void* d_ws, size_t ws_size,
                              hipStream_t stream) {
  (void)in_sizes; (void)n_in; (void)out_size; (void)ws_size;
  const float* z     = (const float*)d_in[0];  // [16,64,256]
  const float* clean = (const float*)d_in[1];  // [64,256]
  const float* Wh    = (const float*)d_in[2];  // [512,512]
  const float* Wg    = (const float*)d_in[3];  // [512,256]
  const float* bh    = (const float*)d_in[4];  // [512]
  const float* lng   = (const float*)d_in[5];  // [512]
  const float* lnb   = (const float*)d_in[6];  // [512]
  const float* alpha = (const float*)d_in[7];  // [1]
  const float* Whead = (const float*)d_in[8];  // [256,512]
  const float* bhead = (const float*)d_in[9];  // [256]
  float* out = (float*)d_out;
  float* ws  = (float*)d_ws;

  float* h_state = ws;                          // 64*512
  float* h_base  = h_state + BSZ * D_H;         // 64*512
  float* hist    = h_base + BSZ * D_H;          // 15*64*512
  float* pred    = hist + 15 * BSZ * D_H;       // 64*256

  for (int t = 0; t < TSTEPS; t++) {
    gemm_base_kernel<<<4, 512, 0, stream>>>(h_state, z + (size_t)t * BSZ * D_G,
                                            Wh, Wg, bh, lng, lnb,
                                            h_base, h_state, t > 0 ? 1 : 0);
    settle_kernel<<<BSZ, 512, 0, stream>>>(h_state, h_base, hist, lng, lnb, alpha,
                                           t, (t == TSTEPS - 1) ? 1 : 0);
  }
  head_kernel<<<4, 512, 0, stream>>>(h_state, Whead, bhead, pred);
  loss_kernel<<<1, 256, 0, stream>>>(pred, clean, out);
}